// ClusterGCNLayer_66563403153821
// MI455X (gfx1250) — compile-verified
//
#include <hip/hip_runtime.h>

typedef float v2f __attribute__((ext_vector_type(2)));
typedef float v8f __attribute__((ext_vector_type(8)));

#define C_DIM 256

// ---------------------------------------------------------------- init pass
__global__ void k_init(float* __restrict__ deg, int* __restrict__ has_edge,
                       int N, int NC) {
    int i = blockIdx.x * blockDim.x + threadIdx.x;
    if (i < N) deg[i] = 1.0f;          // self-loop contribution
    if (i < NC) has_edge[i] = 0;
}

// -------------------------------------------------- degree / has_edge pass
__global__ void k_edges(const int* __restrict__ src, const int* __restrict__ dst,
                        const int* __restrict__ assign,
                        float* __restrict__ deg, int* __restrict__ has_edge, int E) {
    int e = blockIdx.x * blockDim.x + threadIdx.x;
    if (e >= E) return;
    int s = src[e], d = dst[e];
    int cs = assign[s];
    if (cs == assign[d]) {
        unsafeAtomicAdd(&deg[d], 1.0f);   // exact: integer counts < 2^24
        has_edge[cs] = 1;                  // racy all-write-1: fine
    }
}

// ------------------------------------------------------- xW = X @ W (WMMA)
// Wave computes a 16-row x 256-col stripe: 16 accumulator tiles of 16x16,
// K-loop of 64 steps of V_WMMA_F32_16X16X4_F32. Block = 8 waves = 128 rows.
__global__ __launch_bounds__(256) void k_gemm(const float* __restrict__ X,
                                              const float* __restrict__ W,
                                              float* __restrict__ xW, int N) {
    const int lane = threadIdx.x & 31;
    const int wave = threadIdx.x >> 5;
    const int half = lane >> 4;   // 0: K pair {0,1}, 1: K pair {2,3}
    const int m    = lane & 15;
    const int r0   = blockIdx.x * 128 + wave * 16;
    if (r0 >= N) return;          // wave-uniform: EXEC stays all-1s for WMMA

    v8f acc[16] = {};

    // A fragment source: row (r0+m), cols k0 + 2*half + {0,1}  (ISA 32-bit A layout)
    int rowA = r0 + m;
    if (rowA >= N) rowA = N - 1;  // clamp; garbage rows never stored
    const float* __restrict__ Arow = X + (size_t)rowA * C_DIM + 2 * half;
    // B fragment source: W[(k0 + 2*half + {0,1})][t*16 + m]  (row-striped B layout)
    const float* __restrict__ Wb = W + (size_t)(2 * half) * C_DIM + m;

    for (int k0 = 0; k0 < C_DIM; k0 += 4) {
        v2f a;
        a.x = Arow[k0];
        a.y = Arow[k0 + 1];
#pragma unroll
        for (int t = 0; t < 16; ++t) {
            v2f bf;
            bf.x = Wb[(size_t)k0 * C_DIM + t * 16];
            bf.y = Wb[(size_t)(k0 + 1) * C_DIM + t * 16];
            acc[t] = __builtin_amdgcn_wmma_f32_16x16x4_f32(
                false, a, false, bf, (short)0, acc[t], false, false);
        }
    }

    // C/D layout: VGPR v -> row (r0 + v + 8*half), col (t*16 + m)
#pragma unroll
    for (int t = 0; t < 16; ++t) {
#pragma unroll
        for (int v = 0; v < 8; ++v) {
            int row = r0 + v + 8 * half;
            if (row < N) xW[(size_t)row * C_DIM + t * 16 + m] = acc[t][v];
        }
    }
}

// ------------------- out init: self-loop + bias, or passthrough X if cluster
// has no intra edges. Rows in no-edge clusters never receive scatter atomics
// (an intra edge into dst implies dst's cluster has_edge), so no post-pass.
__global__ void k_self(const float* __restrict__ X, const float* __restrict__ xW,
                       const float* __restrict__ deg, const float* __restrict__ b,
                       const int* __restrict__ assign, const int* __restrict__ has_edge,
                       float* __restrict__ out, int N) {
    int idx = blockIdx.x * blockDim.x + threadIdx.x;
    if (idx >= N * C_DIM) return;
    int n = idx >> 8;
    int c = idx & (C_DIM - 1);
    if (has_edge[assign[n]]) {
        out[idx] = xW[idx] / deg[n] + b[c];
    } else {
        out[idx] = X[idx];
    }
}

// ----------------------------- scatter: out[dst] += w_e * xW[src] (atomics)
// One wave per edge; lane handles 8 consecutive channels (2x b128 loads,
// 8x global_atomic_add_f32).
__global__ __launch_bounds__(256) void k_scatter(const int* __restrict__ src,
                                                 const int* __restrict__ dst,
                                                 const int* __restrict__ assign,
                                                 const float* __restrict__ deg,
                                                 const float* __restrict__ xW,
                                                 float* __restrict__ out, int E) {
    int e    = (blockIdx.x * blockDim.x + threadIdx.x) >> 5;
    int lane = threadIdx.x & 31;
    if (e >= E) return;
    int s = src[e], d = dst[e];
    if (assign[s] != assign[d]) return;   // non-intra edge: weight 0
    float w = rsqrtf(deg[s]) * rsqrtf(deg[d]);
    const float4* xs = (const float4*)(xW + (size_t)s * C_DIM) + lane * 2;
    float4 v0 = xs[0];
    float4 v1 = xs[1];
    float* o = out + (size_t)d * C_DIM + lane * 8;
    unsafeAtomicAdd(o + 0, w * v0.x);
    unsafeAtomicAdd(o + 1, w * v0.y);
    unsafeAtomicAdd(o + 2, w * v0.z);
    unsafeAtomicAdd(o + 3, w * v0.w);
    unsafeAtomicAdd(o + 4, w * v1.x);
    unsafeAtomicAdd(o + 5, w * v1.y);
    unsafeAtomicAdd(o + 6, w * v1.z);
    unsafeAtomicAdd(o + 7, w * v1.w);
}

extern "C" void kernel_launch(void* const* d_in, const int* in_sizes, int n_in,
                              void* d_out, int out_size, void* d_ws, size_t ws_size,
                              hipStream_t stream) {
    const float* X      = (const float*)d_in[0];
    const float* W      = (const float*)d_in[1];
    const float* b      = (const float*)d_in[2];
    const int*   assign = (const int*)d_in[3];
    const int*   ei     = (const int*)d_in[4];

    const int N  = in_sizes[0] / C_DIM;   // 50000
    const int E  = in_sizes[4] / 2;       // 300000
    const int NC = 256;                    // >= N_CLUSTERS (100); sized generously

    const int* src = ei;
    const int* dst = ei + E;

    // workspace layout: deg[N] | has_edge[NC] | xW[N*256]
    char* ws = (char*)d_ws;
    size_t off = 0;
    float* deg = (float*)(ws + off);
    off += (((size_t)N * 4) + 255) & ~(size_t)255;
    int* has_edge = (int*)(ws + off);
    off += (((size_t)NC * 4) + 255) & ~(size_t)255;
    float* xW = (float*)(ws + off);

    float* out = (float*)d_out;

    k_init<<<(N + 255) / 256, 256, 0, stream>>>(deg, has_edge, N, NC);
    k_edges<<<(E + 255) / 256, 256, 0, stream>>>(src, dst, assign, deg, has_edge, E);
    k_gemm<<<(N + 127) / 128, 256, 0, stream>>>(X, W, xW, N);
    k_self<<<(N * C_DIM + 255) / 256, 256, 0, stream>>>(X, xW, deg, b, assign,
                                                        has_edge, out, N);
    k_scatter<<<(E + 7) / 8, 256, 0, stream>>>(src, dst, assign, deg, xW, out, E);
}